// QuantumReservoir_62216896250195
// MI455X (gfx1250) — compile-verified
//
#include <hip/hip_runtime.h>
#include <math.h>

typedef __attribute__((ext_vector_type(2))) float v2f;
typedef __attribute__((ext_vector_type(8))) float v8f;

#define NQ        6
#define DIM       64
#define NL        2
#define USTRIDE   65     // padded row stride for U (banks: (r+k)%64)
#define PSTRIDE   36     // padded row stride for P / state
#define BCOLS     32     // batch columns handled per block
#define NTHREADS  128    // 4 wave32 waves

// Apply a 1-qubit complex 2x2 gate on bit position `pos` (left-multiply onto U,
// acting on the row index) across all 64 columns. If withCZ, the CZ-chain
// diagonal sign (applied logically BEFORE this gate) is folded into the loads;
// the pass over bit `pos` touches every row exactly once, so each row gets its
// sign applied exactly once. Each thread owns whole (r0,r1) pairs -> race-free.
__device__ __forceinline__ void apply_gate(float* Ur, float* Ui, int pos, bool withCZ,
    float g00r, float g00i, float g01r, float g01i,
    float g10r, float g10i, float g11r, float g11i, int tid)
{
    const int bit = 1 << pos;
    for (int p = tid; p < 32 * DIM; p += NTHREADS) {
        int col = p & 63;
        int h   = p >> 6;                 // 0..31 pair index
        int low = h & (bit - 1);
        int r0  = ((h ^ low) << 1) | low; // insert 0 at `pos`
        int r1  = r0 | bit;
        int i0 = r0 * USTRIDE + col;
        int i1 = r1 * USTRIDE + col;
        float ar = Ur[i0], ai = Ui[i0];
        float br = Ur[i1], bi = Ui[i1];
        if (withCZ) {                      // wave-uniform branch
            float d0 = (__popc(r0 & (r0 >> 1) & 0x1f) & 1) ? -1.0f : 1.0f;
            float d1 = (__popc(r1 & (r1 >> 1) & 0x1f) & 1) ? -1.0f : 1.0f;
            ar *= d0; ai *= d0;
            br *= d1; bi *= d1;
        }
        float n0r = g00r*ar - g00i*ai + g01r*br - g01i*bi;
        float n0i = g00r*ai + g00i*ar + g01r*bi + g01i*br;
        float n1r = g10r*ar - g10i*ai + g11r*br - g11i*bi;
        float n1i = g10r*ai + g10i*ar + g11r*bi + g11i*br;
        Ur[i0] = n0r; Ui[i0] = n0i;
        Ur[i1] = n1r; Ui[i1] = n1i;
    }
}

__global__ __launch_bounds__(NTHREADS)
void qreservoir_kernel(const float* __restrict__ x, const float* __restrict__ w,
                       float* __restrict__ out, int S)
{
    __shared__ float Ur[DIM * USTRIDE];   // Re(U_s)
    __shared__ float Ui[DIM * USTRIDE];   // Im(U_s)
    __shared__ float P [DIM * PSTRIDE];   // real encoding states (64 x 32)
    __shared__ float Sr[DIM * PSTRIDE];   // Re(result states)
    __shared__ float Si[DIM * PSTRIDE];   // Im(result states)
    __shared__ float cs[BCOLS][NQ];
    __shared__ float sn[BCOLS][NQ];

    const int s   = blockIdx.x;
    const int b0  = blockIdx.y * BCOLS;
    const int tid = threadIdx.x;

    // ---- Phase 0: per-column encoding angles: RY(pi*f)|0> = (cos(pi f/2), sin(pi f/2))
    for (int idx = tid; idx < BCOLS * NQ; idx += NTHREADS) {
        int c = idx / NQ, q = idx % NQ;
        float f = x[((b0 + c) * S + s) * NQ + q];
        float sv, cv;
        sincosf(0.5f * 3.14159265358979323846f * f, &sv, &cv);
        sn[c][q] = sv; cs[c][q] = cv;
    }
    // ---- U = I
    for (int idx = tid; idx < DIM * DIM; idx += NTHREADS) {
        int r = idx >> 6, c = idx & 63;
        Ur[r * USTRIDE + c] = (r == c) ? 1.0f : 0.0f;
        Ui[r * USTRIDE + c] = 0.0f;
    }
    __syncthreads();
    // ---- Phase 1: product-state matrix P (real). qubit q <-> bit (5-q) of amp index.
    for (int idx = tid; idx < DIM * BCOLS; idx += NTHREADS) {
        int i = idx >> 5, c = idx & 31;
        float v = 1.0f;
        #pragma unroll
        for (int q = 0; q < NQ; ++q) {
            int bitv = (i >> (NQ - 1 - q)) & 1;
            v *= bitv ? sn[c][q] : cs[c][q];
        }
        P[i * PSTRIDE + c] = v;
    }
    __syncthreads();

    // ---- Phase 2: build U_s. Per layer: 6 fused gates G_q = Rot_q * RX_q,
    // with the CZ-chain diagonal folded into the q==0 pass (12 sweeps total).
    // Keep the q loop ROLLED: one apply_gate instantiation, compact I$ footprint.
    const float cf = cosf(0.25f);   // cos(FIELD/2), FIELD = 0.5
    const float sf = sinf(0.25f);
    for (int l = 0; l < NL; ++l) {
        #pragma unroll 1
        for (int q = 0; q < NQ; ++q) {
            int base = ((s * NL + l) * NQ + q) * 3;
            float phi = w[base], th = w[base + 1], om = w[base + 2];
            float st, ct; sincosf(0.5f * th, &st, &ct);
            float sa, ca; sincosf(-0.5f * (phi + om), &sa, &ca);  // ep = e^{i a}
            float sb, cb; sincosf( 0.5f * (phi - om), &sb, &cb);  // em = e^{i b}
            // Rot entries
            float r00r =  ca * ct, r00i =  sa * ct;   // ep*ct
            float r01r = -cb * st, r01i = -sb * st;   // -em*st
            float r10r =  cb * st, r10i = -sb * st;   // conj(em)*st
            float r11r =  ca * ct, r11i = -sa * ct;   // conj(ep)*ct
            // G = Rot * RX,  RX = [[cf, -i sf], [-i sf, cf]]
            // (-i sf)*(p + i q) = sf*q - i*sf*p
            float g00r = cf * r00r + sf * r01i;
            float g00i = cf * r00i - sf * r01r;
            float g01r = sf * r00i + cf * r01r;
            float g01i = -sf * r00r + cf * r01i;
            float g10r = cf * r10r + sf * r11i;
            float g10i = cf * r10i - sf * r11r;
            float g11r = sf * r10i + cf * r11r;
            float g11i = -sf * r10r + cf * r11i;
            apply_gate(Ur, Ui, NQ - 1 - q, (q == 0),
                       g00r, g00i, g01r, g01i,
                       g10r, g10i, g11r, g11i, tid);
            __syncthreads();
        }
    }

    // ---- Phase 3: S = U * P via V_WMMA_F32_16X16X4_F32 (two real matmuls).
    // Each wave owns two 16x16 output tiles; K=64 swept as 16 k-steps of 4.
    const int lane = tid & 31;
    const int wv   = tid >> 5;
    const int lh   = lane >> 4;    // half-wave: selects K+{0,1} vs K+{2,3}
    const int lm   = lane & 15;
    #pragma unroll
    for (int t = 0; t < 2; ++t) {
        int T  = wv * 2 + t;
        int m0 = T >> 1;           // 0..3  (row tile)
        int n0 = T & 1;            // 0..1  (col tile)
        v8f accR = {};
        v8f accI = {};
        int rowA = m0 * 16 + lm;
        int colB = n0 * 16 + lm;
        for (int k0 = 0; k0 < 16; ++k0) {
            int kb = k0 * 4 + lh * 2;
            // A (16x4 f32): lanes {0-15,16-31} hold K {0/1, 2/3} in 2 VGPRs
            v2f aR = { Ur[rowA * USTRIDE + kb], Ur[rowA * USTRIDE + kb + 1] };
            v2f aI = { Ui[rowA * USTRIDE + kb], Ui[rowA * USTRIDE + kb + 1] };
            // B (4x16 f32): rows striped across lanes within each VGPR
            v2f bF = { P[kb * PSTRIDE + colB], P[(kb + 1) * PSTRIDE + colB] };
            accR = __builtin_amdgcn_wmma_f32_16x16x4_f32(
                false, aR, false, bF, (short)0, accR, false, false);
            accI = __builtin_amdgcn_wmma_f32_16x16x4_f32(
                false, aI, false, bF, (short)0, accI, false, false);
        }
        #pragma unroll
        for (int v = 0; v < 8; ++v) {
            int row = m0 * 16 + lh * 8 + v;   // C/D layout: VGPR v holds M=v (+8 for hi half)
            Sr[row * PSTRIDE + colB] = accR[v];
            Si[row * PSTRIDE + colB] = accI[v];
        }
    }
    __syncthreads();

    // ---- Phase 4: expectations. 4 lanes per column, 16 amplitudes each.
    {
        int col     = tid >> 2;
        int quarter = tid & 3;
        float e[18];
        #pragma unroll
        for (int k = 0; k < 18; ++k) e[k] = 0.f;
        for (int it = 0; it < 16; ++it) {
            int i = quarter * 16 + ((it + quarter * 4) & 15);  // stagger quarters across banks
            float ar = Sr[i * PSTRIDE + col];
            float ai = Si[i * PSTRIDE + col];
            float prob = ar * ar + ai * ai;
            #pragma unroll
            for (int q = 0; q < NQ; ++q) {
                int pos  = NQ - 1 - q;
                int bitv = (i >> pos) & 1;
                int j    = i ^ (1 << pos);
                float br = Sr[j * PSTRIDE + col];
                float bi = Si[j * PSTRIDE + col];
                e[q]      += ar * br + ai * bi;                  // <X_q>
                float sgn  = bitv ? 1.0f : -1.0f;
                e[6 + q]  += sgn * (ai * br - ar * bi);          // <Y_q>
                e[12 + q] += bitv ? -prob : prob;                // <Z_q>
            }
        }
        #pragma unroll
        for (int k = 0; k < 18; ++k) {                           // reduce 4 lanes (wave32)
            e[k] += __shfl_xor(e[k], 1);
            e[k] += __shfl_xor(e[k], 2);
        }
        if (quarter == 0) {
            int b = b0 + col;
            float* o = out + ((size_t)b * S + s) * 18;
            #pragma unroll
            for (int k = 0; k < 18; ++k) o[k] = e[k];
        }
    }
}

extern "C" void kernel_launch(void* const* d_in, const int* in_sizes, int n_in,
                              void* d_out, int out_size, void* d_ws, size_t ws_size,
                              hipStream_t stream) {
    const float* x = (const float*)d_in[0];   // (B, S, 6) f32
    const float* w = (const float*)d_in[1];   // (S, 2, 6, 3) f32
    float* out = (float*)d_out;               // (B, S, 18) f32
    (void)d_ws; (void)ws_size; (void)n_in; (void)out_size;
    int S = in_sizes[1] / (NL * NQ * 3);      // 512
    int B = in_sizes[0] / (S * NQ);           // 128
    dim3 grid(S, B / BCOLS);                  // (512, 4) workgroups
    qreservoir_kernel<<<grid, NTHREADS, 0, stream>>>(x, w, out, S);
}